// LengthRegulator_83013127897304
// MI455X (gfx1250) — compile-verified
//
#include <hip/hip_runtime.h>

typedef float v2f __attribute__((ext_vector_type(2)));
typedef float v8f __attribute__((ext_vector_type(8)));

#define B_    32
#define T_    512
#define TP_   (T_ + 2)       // padded time axis (zero row before and after)
#define D_    256
#define F_    256
#define KW    3
#define MTOT  (B_ * T_)      // 16384
#define MEL   2048

// ---------------------------------------------------------------------------
// Build zero-padded input: xpad[b, tt, :] = (1 <= tt <= T) ? x[b, tt-1, :] : 0
// ---------------------------------------------------------------------------
__global__ __launch_bounds__(256)
void build_xpad(const float* __restrict__ x, float* __restrict__ xpad)
{
    const int id = blockIdx.x * 256 + threadIdx.x;      // one float4 each
    const int ROW4 = D_ / 4;                            // 64
    const int b  = id / (TP_ * ROW4);
    const int r  = id % (TP_ * ROW4);
    const int tt = r / ROW4;
    const int c4 = r % ROW4;
    float4 v = make_float4(0.f, 0.f, 0.f, 0.f);
    if (tt >= 1 && tt <= T_)
        v = ((const float4*)x)[((size_t)b * T_ + (tt - 1)) * ROW4 + c4];
    ((float4*)xpad)[id] = v;
}

// Zero the 2 pad rows of each batch in a padded [B, TP, F] buffer.
__global__ void zero_pad_rows(float* __restrict__ h)
{
    const int id    = blockIdx.x * 256 + threadIdx.x;   // 64 rows * 256 floats
    const int b     = id >> 9;
    const int which = (id >> 8) & 1;
    const int d     = id & 255;
    h[((size_t)b * TP_ + which * (TP_ - 1)) * F_ + d] = 0.f;
}

// Repack conv weights [F, D, KW] -> wT [KW, D, F] so B-fragment loads are
// lane-contiguous in f.
__global__ __launch_bounds__(256)
void repack_w(const float* __restrict__ w, float* __restrict__ wT)
{
    const int id  = blockIdx.x * 256 + threadIdx.x;     // F*D*KW = 196608
    const int f   = id / (D_ * KW);
    const int r   = id % (D_ * KW);
    const int d   = r / KW;
    const int tap = r % KW;
    wT[((size_t)tap * D_ + d) * F_ + f] = w[id];
}

// ---------------------------------------------------------------------------
// Conv1d(K=3,pad=1) as branchless implicit-im2row GEMM on padded input.
//   out[M = B*T, N = F],  K = KW*D,  via V_WMMA_F32_16X16X4_F32.
// One wave -> 64(M) x 16(N) tile: four 16x16 accumulators, B fragment
// reused across the four M-subtiles. opad selects padded output layout.
// ---------------------------------------------------------------------------
__global__ __launch_bounds__(256)
void conv_gemm_wmma(const float* __restrict__ xpad,  // [B, TP, D]
                    const float* __restrict__ wT,    // [KW, D, F]
                    const float* __restrict__ bias,  // [F]
                    float* __restrict__ out,         // [B, T(+2*opad), F]
                    const int opad)
{
    const int lane = threadIdx.x;        // wave32
    const int l15  = lane & 15;
    const int hi   = lane >> 4;          // 0|1
    const int koff = hi << 1;            // K sub-offset 0|2

    const int gw    = blockIdx.x * 8 + threadIdx.y;
    const int mtile = gw >> 4;           // 256 x 64-row tiles
    const int ntile = gw & 15;           // 16 x 16-col tiles
    const int mbase = mtile << 6;
    const int batch = mbase >> 9;        // 8 tiles per batch (64 | 512)
    const int tloc  = mbase & (T_ - 1);  // local t base of tile
    const float* xb = xpad + (size_t)batch * TP_ * D_;
    const int ncol  = (ntile << 4) + l15;

    v8f acc[4] = { {}, {}, {}, {} };

#pragma unroll
    for (int tap = 0; tap < KW; ++tap) {
        // padded row index = tloc + l15 + s*16 + tap, always in [0, TP)
        const float* a0 = xb + (size_t)(tloc + l15 + tap) * D_ + koff;
        const float* a1 = a0 + 16 * D_;
        const float* a2 = a0 + 32 * D_;
        const float* a3 = a0 + 48 * D_;
        const float* wp = wT + ((size_t)tap * D_ + koff) * F_ + ncol;

#pragma unroll 4
        for (int d0 = 0; d0 < D_; d0 += 4) {
            v2f bf;
            bf.x = wp[(size_t)d0 * F_];
            bf.y = wp[(size_t)d0 * F_ + F_];
            const v2f af0 = *(const v2f*)(a0 + d0);
            const v2f af1 = *(const v2f*)(a1 + d0);
            const v2f af2 = *(const v2f*)(a2 + d0);
            const v2f af3 = *(const v2f*)(a3 + d0);
            acc[0] = __builtin_amdgcn_wmma_f32_16x16x4_f32(false, af0, false, bf, (short)0, acc[0], false, false);
            acc[1] = __builtin_amdgcn_wmma_f32_16x16x4_f32(false, af1, false, bf, (short)0, acc[1], false, false);
            acc[2] = __builtin_amdgcn_wmma_f32_16x16x4_f32(false, af2, false, bf, (short)0, acc[2], false, false);
            acc[3] = __builtin_amdgcn_wmma_f32_16x16x4_f32(false, af3, false, bf, (short)0, acc[3], false, false);
        }
    }

    // C/D layout: VGPR v <-> row v + 8*hi of the 16x16 tile, col = l15.
    const float bcol  = bias[ncol];
    const int   obase = batch * (T_ + 2 * opad) + opad + tloc;
#pragma unroll
    for (int s = 0; s < 4; ++s) {
#pragma unroll
        for (int v = 0; v < 8; ++v)
            out[(size_t)(obase + (s << 4) + v + (hi << 3)) * F_ + ncol] = acc[s][v] + bcol;
    }
}

// ---------------------------------------------------------------------------
// In-place LayerNorm(256) + ReLU over valid rows; pad-aware layout.
// One wave per row, 8 features per lane, __shfl_xor wave32 reduction.
// ---------------------------------------------------------------------------
__global__ __launch_bounds__(256)
void ln_relu_inplace(float* __restrict__ h,
                     const float* __restrict__ g,
                     const float* __restrict__ b,
                     const int pad)
{
    const int row   = blockIdx.x * 8 + threadIdx.y;     // 0..MTOT-1
    const int batch = row >> 9;
    const int t     = row & (T_ - 1);
    const int lane  = threadIdx.x;
    float* p = h + ((size_t)batch * (T_ + 2 * pad) + pad + t) * F_;

    float v[8];
    float s = 0.f, ss = 0.f;
#pragma unroll
    for (int i = 0; i < 8; ++i) {
        v[i] = p[lane + 32 * i];
        s  += v[i];
        ss += v[i] * v[i];
    }
#pragma unroll
    for (int m = 16; m > 0; m >>= 1) {
        s  += __shfl_xor(s,  m, 32);
        ss += __shfl_xor(ss, m, 32);
    }
    const float mean = s * (1.f / F_);
    const float var  = ss * (1.f / F_) - mean * mean;
    const float inv  = rsqrtf(var + 1e-5f);
#pragma unroll
    for (int i = 0; i < 8; ++i) {
        const int f = lane + 32 * i;
        const float y = (v[i] - mean) * inv * g[f] + b[f];
        p[f] = fmaxf(y, 0.f);
    }
}

// ---------------------------------------------------------------------------
// Linear head: dp[row] = dot(h[row,:], lin_w) + lin_b (h unpadded).
// ---------------------------------------------------------------------------
__global__ __launch_bounds__(256)
void linear_head(const float* __restrict__ h,
                 const float* __restrict__ lw,
                 const float* __restrict__ lb,
                 float* __restrict__ dp)
{
    const int row  = blockIdx.x * 8 + threadIdx.y;
    const int lane = threadIdx.x;
    const float* p = h + (size_t)row * F_;
    float s = 0.f;
#pragma unroll
    for (int i = 0; i < 8; ++i) {
        const int f = lane + 32 * i;
        s += p[f] * lw[f];
    }
#pragma unroll
    for (int m = 16; m > 0; m >>= 1)
        s += __shfl_xor(s, m, 32);
    if (lane == 0) dp[row] = s + lb[0];
}

// ---------------------------------------------------------------------------
// Per-batch inclusive cumsum of durations (32 batches, serial 512 each).
// ---------------------------------------------------------------------------
__global__ void cumsum_k(const int* __restrict__ target, int* __restrict__ ends)
{
    const int b = threadIdx.x;   // 0..31
    int s = 0;
    for (int t = 0; t < T_; ++t) {
        s += target[b * T_ + t];
        ends[b * T_ + t] = s;
    }
}

// ---------------------------------------------------------------------------
// Length regulator: interval-indicator einsum == row gather.
// out[b, m, :] = x[b, first t with ends[t] > m, :]  (zero past total length).
// ---------------------------------------------------------------------------
__global__ __launch_bounds__(256)
void gather_lr(const float* __restrict__ x,
               const int* __restrict__ ends,
               float* __restrict__ out)
{
    const int b      = blockIdx.x;
    const int m      = blockIdx.y * 4 + (threadIdx.x >> 6);
    const int lane64 = threadIdx.x & 63;
    const int* e     = ends + b * T_;
    const int total  = e[T_ - 1];

    float4 val = make_float4(0.f, 0.f, 0.f, 0.f);
    if (m < total) {
        int lo = 0, hiidx = T_ - 1;
        while (lo < hiidx) {
            const int mid = (lo + hiidx) >> 1;
            if (e[mid] > m) hiidx = mid; else lo = mid + 1;
        }
        val = ((const float4*)(x + ((size_t)b * T_ + lo) * D_))[lane64];
    }
    ((float4*)(out + ((size_t)b * MEL + m) * D_))[lane64] = val;
}

// ---------------------------------------------------------------------------
extern "C" void kernel_launch(void* const* d_in, const int* in_sizes, int n_in,
                              void* d_out, int out_size, void* d_ws, size_t ws_size,
                              hipStream_t stream)
{
    const float* x    = (const float*)d_in[0];
    const float* c1w  = (const float*)d_in[1];
    const float* c1b  = (const float*)d_in[2];
    const float* g1   = (const float*)d_in[3];
    const float* b1   = (const float*)d_in[4];
    const float* c2w  = (const float*)d_in[5];
    const float* c2b  = (const float*)d_in[6];
    const float* g2   = (const float*)d_in[7];
    const float* b2   = (const float*)d_in[8];
    const float* lw   = (const float*)d_in[9];
    const float* lb   = (const float*)d_in[10];
    const int*   tgt  = (const int*)d_in[11];

    float* out_lr = (float*)d_out;                      // [B, MEL, D]
    float* dp     = out_lr + (size_t)B_ * MEL * D_;     // [B, T]

    float* xpad  = (float*)d_ws;                        // [B, TP, D]
    float* h1pad = xpad  + (size_t)B_ * TP_ * D_;       // [B, TP, F]
    float* h2    = h1pad + (size_t)B_ * TP_ * F_;       // [B*T, F]
    float* w1T   = h2    + (size_t)MTOT * F_;           // [KW, D, F]
    float* w2T   = w1T   + (size_t)KW * D_ * F_;
    int*   ends  = (int*)(w2T + (size_t)KW * D_ * F_);  // [B, T]

    dim3 blk(32, 8);
    const int padBlocks = (B_ * TP_ * (D_ / 4) + 255) / 256;   // float4 copy
    const int wBlocks   = (F_ * D_ * KW + 255) / 256;

    // Prologue: padding + weight repack (tiny, L2-resident)
    build_xpad<<<padBlocks, 256, 0, stream>>>(x, xpad);
    zero_pad_rows<<<64, 256, 0, stream>>>(h1pad);
    repack_w<<<wBlocks, 256, 0, stream>>>(c1w, w1T);
    repack_w<<<wBlocks, 256, 0, stream>>>(c2w, w2T);

    // Duration-predictor path: branchless fp32 WMMA GEMM convs + LN/ReLU
    conv_gemm_wmma<<<512, blk, 0, stream>>>(xpad,  w1T, c1b, h1pad, 1);
    ln_relu_inplace<<<2048, blk, 0, stream>>>(h1pad, g1, b1, 1);
    conv_gemm_wmma<<<512, blk, 0, stream>>>(h1pad, w2T, c2b, h2, 0);
    ln_relu_inplace<<<2048, blk, 0, stream>>>(h2, g2, b2, 0);
    linear_head<<<2048, blk, 0, stream>>>(h2, lw, lb, dp);

    // Length regulator: bandwidth-bound gather (HBM roofline ~128 MB)
    cumsum_k<<<1, 32, 0, stream>>>(tgt, ends);
    gather_lr<<<dim3(B_, MEL / 4), 256, 0, stream>>>(x, ends, out_lr);
}